// SVDHead_31782757990755
// MI455X (gfx1250) — compile-verified
//
#include <hip/hip_runtime.h>

// Live computation of the reference (everything else is dead code):
//   r  = eye(3)                          -> d_out[0..8], row-major
//   tb = -mean(mkpts0, axis=1)           -> d_out[9..11]
// mkpts0 is [3, 8192] float32, row-major.

typedef __attribute__((ext_vector_type(2))) float v2f;
typedef __attribute__((ext_vector_type(8))) float v8f;

#define NPTS 8192

__global__ __launch_bounds__(96)
void svdhead_identity_tb_kernel(const float* __restrict__ mkpts0,
                                float* __restrict__ out)
{
    const int tid  = threadIdx.x;
    const int wave = tid >> 5;   // wave 0..2 -> row 0..2 of mkpts0
    const int lane = tid & 31;

    __shared__ float partial[96];

    // WMMA ones-matmul reduction over one row of mkpts0.
    // Per iteration: load a contiguous 64-float chunk into the wave's A-tile
    // (2 f32 per lane = 16x4 f32 A operand) and do D = A * ones(4x16) + C.
    // With B == all-ones, every A element contributes once to each of D's 16
    // identical columns, so sum(acc over lanes+VGPRs) == 16 * sum(chunk),
    // independent of the exact (M,K) <-> lane mapping.
    const float* row = mkpts0 + wave * NPTS;

    v2f ones;
    ones.x = 1.0f;
    ones.y = 1.0f;
    v8f acc = {};

    for (int c = 0; c < NPTS / 64; ++c) {
        const float* p = row + c * 64 + lane * 2;
        __builtin_prefetch(p + 256, 0, 1);   // global_prefetch_b8 (gfx1250)
        v2f a;
        a.x = p[0];
        a.y = p[1];
        // 8 args: (neg_a, A, neg_b, B, c_mod, C, reuse_a, reuse_b)
        acc = __builtin_amdgcn_wmma_f32_16x16x4_f32(
            false, a, false, ones, (short)0, acc, false, false);
    }

    float lane_sum = acc[0] + acc[1] + acc[2] + acc[3] +
                     acc[4] + acc[5] + acc[6] + acc[7];
    partial[tid] = lane_sum;
    __syncthreads();

    if (lane == 0) {
        float s = 0.0f;
        for (int i = 0; i < 32; ++i) s += partial[wave * 32 + i];
        s *= (1.0f / 16.0f);                  // 16 identical D columns
        out[9 + wave] = -(s / (float)NPTS);   // tb = -mean(mkpts0, axis=1)
    }
    if (tid < 9) {
        out[tid] = ((tid & 3) == 0) ? 1.0f : 0.0f;  // r = I3, row-major
    }
}

extern "C" void kernel_launch(void* const* d_in, const int* in_sizes, int n_in,
                              void* d_out, int out_size, void* d_ws, size_t ws_size,
                              hipStream_t stream)
{
    (void)in_sizes; (void)n_in; (void)d_ws; (void)ws_size; (void)out_size;
    const float* mkpts0 = (const float*)d_in[0];   // [3, 8192] f32
    // d_in[1] (mkpts1) and d_in[2] (valid_scores) are dead in the reference
    // output: r is overwritten with identity, tb depends only on mkpts0.
    float* out = (float*)d_out;                    // 12 floats: r (9) + tb (3)
    svdhead_identity_tb_kernel<<<1, 96, 0, stream>>>(mkpts0, out);
}